// BiMamba2DSlow1_31971736552200
// MI455X (gfx1250) — compile-verified
//
#include <hip/hip_runtime.h>
#include <cmath>
#include <cstdint>

typedef __attribute__((ext_vector_type(2))) float v2f;
typedef __attribute__((ext_vector_type(8))) float v8f;
typedef __attribute__((ext_vector_type(4))) unsigned int v4u;
typedef __attribute__((ext_vector_type(8))) int v8i;
typedef __attribute__((ext_vector_type(4))) int v4i;

// Problem sizes (compile-time constants from the reference)
constexpr int Bn    = 4;
constexpr int Hh    = 64, Ww = 64;
constexpr int Lseq  = Hh * Ww;        // 4096
constexpr int DM    = 96;             // d_model
constexpr int DI    = 192;            // d_inner
constexpr int DSt   = 16;             // d_state
constexpr int DTR   = 6;              // dt_rank
constexpr int CDBL  = DTR + 2 * DSt;  // 38
constexpr int MPIX  = Bn * Lseq;      // 16384 pixels
constexpr int NXZ   = 2 * DI;         // 384
constexpr int NCHUNK = 32;            // scan chunks
constexpr int CLEN   = Lseq / NCHUNK; // 128 steps per chunk

// direction->canonical pixel permutation (dir&1: wh-major, dir&2: reversed)
__device__ __forceinline__ int perm_pixel(int dir, int t) {
  int tt = (dir & 2) ? (Lseq - 1 - t) : t;
  return (dir & 1) ? (((tt & 63) << 6) | (tt >> 6)) : tt;
}

// ---------------------------------------------------------------------------
// K1: xz = x @ W_in^T   (M=16384, K=96, N=384), fp32 WMMA 16x16x4
// ---------------------------------------------------------------------------
__global__ __launch_bounds__(256) void k_gemm_in(const float* __restrict__ x,
                                                 const float* __restrict__ w_in,
                                                 float* __restrict__ xz) {
  const int wid  = blockIdx.x * (blockDim.x >> 5) + (threadIdx.x >> 5);
  const int lane = threadIdx.x & 31;
  const int hf   = lane >> 4;
  const int ln   = lane & 15;
  const int ntn  = NXZ / 16;                 // 24 N-tiles
  const int tm   = wid / ntn;
  const int tn   = wid % ntn;
  const int m    = tm * 16 + ln;
  const int n    = tn * 16 + ln;
  v8f c = {0.f, 0.f, 0.f, 0.f, 0.f, 0.f, 0.f, 0.f};
  const float* arow = x + (size_t)m * DM;
  const float* brow = w_in + (size_t)n * DM;  // W_in is (N=384, K=96) row-major
  for (int k0 = 0; k0 < DM; k0 += 4) {
    const int kk = k0 + 2 * hf;
    v2f a = *(const v2f*)(arow + kk);
    v2f b = *(const v2f*)(brow + kk);
    c = __builtin_amdgcn_wmma_f32_16x16x4_f32(false, a, false, b, (short)0, c,
                                              false, false);
  }
  const int rbase = tm * 16 + 8 * hf;
#pragma unroll
  for (int i = 0; i < 8; ++i)
    xz[(size_t)(rbase + i) * NXZ + tn * 16 + ln] = c[i];
}

// ---------------------------------------------------------------------------
// K2a: one-shot conv-weight transpose to [ky*kx][ci][oc] for coalesced
//      B-fragment loads in the implicit-GEMM conv.
// ---------------------------------------------------------------------------
__global__ __launch_bounds__(256) void k_wt(const float* __restrict__ cw,
                                            float* __restrict__ wt) {
  const int e    = blockIdx.x * blockDim.x + threadIdx.x;  // 9*192*192 exact
  const int oc   = e % DI;
  const int ci   = (e / DI) % DI;
  const int kidx = e / (DI * DI);
  wt[e] = cw[(size_t)(oc * DI + ci) * 9 + kidx];
}

// ---------------------------------------------------------------------------
// K2b: 3x3 conv 192->192 (pad 1) + bias + SiLU, implicit GEMM on fp32 WMMA.
//      Each wave computes a 16x32 output tile (2 accumulators share A-frag).
// ---------------------------------------------------------------------------
__global__ __launch_bounds__(256) void k_conv(const float* __restrict__ xz,
                                              const float* __restrict__ wt,
                                              const float* __restrict__ cb,
                                              float* __restrict__ xc) {
  const int wid  = blockIdx.x * (blockDim.x >> 5) + (threadIdx.x >> 5);
  const int lane = threadIdx.x & 31;
  const int hf   = lane >> 4;
  const int ln   = lane & 15;
  const int ntn  = DI / 32;                  // 6 double-N-tiles
  const int tm   = wid / ntn;
  const int tn   = wid % ntn;
  const int m    = tm * 16 + ln;             // global pixel (A row)
  const int bimg = m >> 12;
  const int l    = m & (Lseq - 1);
  const int hp   = l >> 6;
  const int wp   = l & 63;
  const int n0   = tn * 32 + ln;             // output channels (B columns)
  const int n1   = n0 + 16;
  v8f c0 = {0.f, 0.f, 0.f, 0.f, 0.f, 0.f, 0.f, 0.f};
  v8f c1 = {0.f, 0.f, 0.f, 0.f, 0.f, 0.f, 0.f, 0.f};
  for (int dy = 0; dy < 3; ++dy) {
    const int hh = hp + dy - 1;
    for (int dx = 0; dx < 3; ++dx) {
      const int wq = wp + dx - 1;
      const bool valid = (hh >= 0) & (hh < Hh) & (wq >= 0) & (wq < Ww);
      const float* aptr =
          xz + (size_t)((bimg << 12) + (hh << 6) + wq) * NXZ;  // xp stride 384
      const float* wsl = wt + (size_t)(dy * 3 + dx) * DI * DI; // [ci][oc]
      for (int k0 = 0; k0 < DI; k0 += 4) {
        const int kk = k0 + 2 * hf;
        v2f a;
        if (valid) {
          a = *(const v2f*)(aptr + kk);
        } else {
          a[0] = 0.f; a[1] = 0.f;
        }
        v2f b0, b1;
        b0[0] = wsl[(size_t)kk * DI + n0];        // lane-contiguous
        b0[1] = wsl[(size_t)(kk + 1) * DI + n0];
        b1[0] = wsl[(size_t)kk * DI + n1];
        b1[1] = wsl[(size_t)(kk + 1) * DI + n1];
        c0 = __builtin_amdgcn_wmma_f32_16x16x4_f32(false, a, false, b0,
                                                   (short)0, c0, false, false);
        c1 = __builtin_amdgcn_wmma_f32_16x16x4_f32(false, a, false, b1,
                                                   (short)0, c1, false, false);
      }
    }
  }
  const float bias0 = cb[n0];
  const float bias1 = cb[n1];
  const int   rbase = tm * 16 + 8 * hf;
#pragma unroll
  for (int i = 0; i < 8; ++i) {
    float v0 = c0[i] + bias0;
    v0 = v0 / (1.f + __expf(-v0));           // SiLU
    float v1 = c1[i] + bias1;
    v1 = v1 / (1.f + __expf(-v1));
    xc[(size_t)(rbase + i) * DI + n0] = v0;
    xc[(size_t)(rbase + i) * DI + n1] = v1;
  }
}

// ---------------------------------------------------------------------------
// K3: x_dbl = x_proj_w @ xs, as WMMA GEMM. M = 16*4096 sequence elems,
//     N = 48 (38 padded to 3 tiles), K = 192. Store masked past col 38.
// ---------------------------------------------------------------------------
__global__ __launch_bounds__(256) void k_xdbl(const float* __restrict__ xc,
                                              const float* __restrict__ xpw,
                                              float* __restrict__ xdbl) {
  const int wid  = blockIdx.x * (blockDim.x >> 5) + (threadIdx.x >> 5);
  const int lane = threadIdx.x & 31;
  const int hf   = lane >> 4;
  const int ln   = lane & 15;
  const int tm   = wid / 3;                  // 4096 row tiles
  const int tn   = wid % 3;                  // 3 col tiles (48 cols)
  const int rowg = tm * 16 + ln;             // = bdir*4096 + t
  const int bdir = rowg >> 12;
  const int t    = rowg & (Lseq - 1);
  const int dir  = bdir >> 2;
  const int b    = bdir & 3;
  const int p    = perm_pixel(dir, t);
  const float* arow = xc + (size_t)((b << 12) + p) * DI;
  const int n  = tn * 16 + ln;
  const int nc = (n < CDBL) ? n : (CDBL - 1);      // clamp to avoid OOB loads
  const float* brow = xpw + (size_t)nc * DI;       // x_proj_w (38,192) row-major
  v8f c = {0.f, 0.f, 0.f, 0.f, 0.f, 0.f, 0.f, 0.f};
  for (int k0 = 0; k0 < DI; k0 += 4) {
    const int kk = k0 + 2 * hf;
    v2f a = *(const v2f*)(arow + kk);
    v2f b2 = *(const v2f*)(brow + kk);
    c = __builtin_amdgcn_wmma_f32_16x16x4_f32(false, a, false, b2, (short)0, c,
                                              false, false);
  }
  const int rbase = tm * 16 + 8 * hf;
  const int col   = tn * 16 + ln;
  if (col < CDBL) {
#pragma unroll
    for (int i = 0; i < 8; ++i)
      xdbl[(size_t)(rbase + i) * CDBL + col] = c[i];
  }
}

// ---------------------------------------------------------------------------
// K4: chunked selective scan. Block = 16 ch x 16 states of one
//     (dir,batch,chunk). The chunk's x_dbl slab (128x38 fp32 = 19 KB,
//     contiguous) is staged into LDS by the Tensor Data Mover: wave 0 builds
//     a D# (1-D tile, 4864 dwords) and issues TENSOR_LOAD_TO_LDS, waits on
//     TENSORcnt, then the workgroup barrier publishes the LDS slab.
//     u values (128x16, gathered) are staged by normal loads.
//     PASS 0: local scan from h=0, record (prod dA, h_end) per chunk.
//     PASS 1: re-scan seeded with corrected h_in, emit y to canonical pixels.
// ---------------------------------------------------------------------------
template <int PASS>
__global__ __launch_bounds__(256) void k_scan_chunk(
    const float* __restrict__ xc, const float* __restrict__ xdbl,
    const float* __restrict__ alog, const float* __restrict__ dtw,
    const float* __restrict__ dtb, const float* __restrict__ Dsk,
    const float* __restrict__ hin, float* __restrict__ hprod,
    float* __restrict__ hend, float* __restrict__ y4) {
  const int ngrp  = DI / 16;                       // 12 channel groups
  const int bx    = blockIdx.x;
  const int bdir  = bx / (ngrp * NCHUNK);
  const int rem   = bx % (ngrp * NCHUNK);
  const int cg    = rem / NCHUNK;
  const int chunk = rem % NCHUNK;
  const int dir   = bdir >> 2;
  const int b     = bdir & 3;
  const int tid   = threadIdx.x;
  const int dch   = tid >> 4;                      // channel-in-group
  const int d     = cg * 16 + dch;
  const int s     = tid & 15;

  const float Aval = -__expf(alog[d * DSt + s]);
  float wv[DTR];
#pragma unroll
  for (int r = 0; r < DTR; ++r) wv[r] = dtw[d * DTR + r];
  const float dtbv = dtb[d];
  const float Dv   = Dsk[d];

  __shared__ float sx[CLEN * CDBL];                // 128*38 floats = 19 KB
  __shared__ float su[CLEN * 16];                  // 128*16 floats = 8 KB
  const int t0 = chunk * CLEN;

  // --- TDM: DMA the contiguous x_dbl slab into LDS (wave 0 only; uniform
  // branch so other waves skip the instruction -- TDM ignores EXEC).
  const float* src = xdbl + (size_t)(bdir * Lseq + t0) * CDBL;
  if (tid < 32) {
    const uint64_t ga   = (uint64_t)(uintptr_t)src;
    const uint32_t lds  = (uint32_t)(uintptr_t)(&sx[0]);  // low 32 = LDS offset
    const uint32_t ndw  = (uint32_t)(CLEN * CDBL);        // 4864 dwords
    v4u g0;
    g0[0] = 1u;                                            // count=1, load
    g0[1] = lds;                                           // lds_addr
    g0[2] = (uint32_t)ga;                                  // global_addr lo
    g0[3] = (uint32_t)((ga >> 32) & 0x01FFFFFFu) | (2u << 30);  // hi | type=2
    v8i g1;
    g1[0] = (int)(2u << 16);                 // wg_mask=0, data_size=4B
    g1[1] = (int)((ndw & 0xFFFFu) << 16);    // tensor_dim0[15:0]
    g1[2] = (int)((ndw >> 16) | (1u << 16)); // tensor_dim0[31:16] | dim1=1
    g1[3] = (int)((ndw & 0xFFFFu) << 16);    // dim1 hi=0 | tile_dim0
    g1[4] = 1;                               // tile_dim1=1, tile_dim2=0
    g1[5] = (int)ndw;                        // tensor_dim0_stride lo
    g1[6] = 0;
    g1[7] = 0;
    v4i gz = {0, 0, 0, 0};
#if __clang_major__ >= 23
    v8i gz8 = {0, 0, 0, 0, 0, 0, 0, 0};
    __builtin_amdgcn_tensor_load_to_lds(g0, g1, gz, gz, gz8, 0);
#else
    __builtin_amdgcn_tensor_load_to_lds(g0, g1, gz, gz, 0);
#endif
    __builtin_amdgcn_s_wait_tensorcnt(0);
  }
  // Stage gathered u values with normal loads.
  for (int e = tid; e < CLEN * 16; e += 256) {
    const int tl = e >> 4;
    const int dc = e & 15;
    const int p  = perm_pixel(dir, t0 + tl);
    su[e] = xc[(size_t)((b << 12) + p) * DI + cg * 16 + dc];
  }
  __syncthreads();

  const size_t hidx =
      (((size_t)bdir * DI + d) * NCHUNK + chunk) * DSt + s;
  float h  = (PASS == 0) ? 0.f : hin[hidx];
  float pr = 1.f;

  for (int tl = 0; tl < CLEN; ++tl) {
    const float* xr = sx + tl * CDBL;
    float dl = dtbv;
#pragma unroll
    for (int r = 0; r < DTR; ++r) dl = fmaf(xr[r], wv[r], dl);
    const float delta = (dl > 20.f) ? dl : log1pf(__expf(dl));  // softplus
    const float u  = su[(tl << 4) | dch];
    const float dA = __expf(delta * Aval);
    h = dA * h + (delta * u) * xr[DTR + s];
    if (PASS == 0) {
      pr *= dA;
    } else {
      float y = h * xr[DTR + DSt + s];
      y += __shfl_xor(y, 1, 32);
      y += __shfl_xor(y, 2, 32);
      y += __shfl_xor(y, 4, 32);
      y += __shfl_xor(y, 8, 32);
      if (s == 0) {
        const int p = perm_pixel(dir, t0 + tl);
        y4[((size_t)bdir * Lseq + p) * DI + d] = y + Dv * u;
      }
    }
  }
  if (PASS == 0) {
    hprod[hidx] = pr;
    hend[hidx]  = h;
  }
}

// ---------------------------------------------------------------------------
// K4b: serial combine of the 32 chunk summaries per (bdir,d,s):
//      h_in[c] = running;  running = prod[c]*running + end[c]
// ---------------------------------------------------------------------------
__global__ __launch_bounds__(256) void k_scan_mid(const float* __restrict__ hprod,
                                                  const float* __restrict__ hend,
                                                  float* __restrict__ hin) {
  const int gid  = blockIdx.x * blockDim.x + threadIdx.x;  // 16*192*16 exact
  const int s    = gid & 15;
  const int rest = gid >> 4;                               // bdir*DI + d
  const size_t base = (size_t)rest * NCHUNK * DSt + s;
  float hr = 0.f;
  for (int c = 0; c < NCHUNK; ++c) {
    const size_t idx = base + (size_t)c * DSt;
    if (c + 4 < NCHUNK) {
      __builtin_prefetch(hprod + idx + 4 * DSt, 0, 1);
      __builtin_prefetch(hend + idx + 4 * DSt, 0, 1);
    }
    hin[idx] = hr;
    hr = fmaf(hprod[idx], hr, hend[idx]);
  }
}

// ---------------------------------------------------------------------------
// K5: yy[m,d] = (sum over 4 dirs of y4) * SiLU(z[m,d]);  z = xz[:,192:384]
// ---------------------------------------------------------------------------
__global__ __launch_bounds__(256) void k_combine(const float* __restrict__ y4,
                                                 const float* __restrict__ xz,
                                                 float* __restrict__ yy) {
  const size_t e = (size_t)blockIdx.x * blockDim.x + threadIdx.x;
  if (e >= (size_t)MPIX * DI) return;
  const size_t m = e / DI;
  const int    d = (int)(e % DI);
  float sum = 0.f;
#pragma unroll
  for (int dir = 0; dir < 4; ++dir)
    sum += y4[((size_t)dir * MPIX + m) * DI + d];
  const float z = xz[m * NXZ + DI + d];
  yy[e] = sum * (z / (1.f + __expf(-z)));
}

// ---------------------------------------------------------------------------
// K6: out = yy @ W_out^T   (M=16384, K=192, N=96), fp32 WMMA 16x16x4
// ---------------------------------------------------------------------------
__global__ __launch_bounds__(256) void k_gemm_out(const float* __restrict__ yy,
                                                  const float* __restrict__ wout,
                                                  float* __restrict__ out) {
  const int wid  = blockIdx.x * (blockDim.x >> 5) + (threadIdx.x >> 5);
  const int lane = threadIdx.x & 31;
  const int hf   = lane >> 4;
  const int ln   = lane & 15;
  const int ntn  = DM / 16;                  // 6 N-tiles
  const int tm   = wid / ntn;
  const int tn   = wid % ntn;
  const int m    = tm * 16 + ln;
  const int n    = tn * 16 + ln;
  v8f c = {0.f, 0.f, 0.f, 0.f, 0.f, 0.f, 0.f, 0.f};
  const float* arow = yy + (size_t)m * DI;
  const float* brow = wout + (size_t)n * DI;  // W_out is (96,192) row-major
  for (int k0 = 0; k0 < DI; k0 += 4) {
    const int kk = k0 + 2 * hf;
    v2f a = *(const v2f*)(arow + kk);
    v2f b = *(const v2f*)(brow + kk);
    c = __builtin_amdgcn_wmma_f32_16x16x4_f32(false, a, false, b, (short)0, c,
                                              false, false);
  }
  const int rbase = tm * 16 + 8 * hf;
#pragma unroll
  for (int i = 0; i < 8; ++i)
    out[(size_t)(rbase + i) * DM + tn * 16 + ln] = c[i];
}

// ---------------------------------------------------------------------------
extern "C" void kernel_launch(void* const* d_in, const int* in_sizes, int n_in,
                              void* d_out, int out_size, void* d_ws,
                              size_t ws_size, hipStream_t stream) {
  const float* x    = (const float*)d_in[0];
  const float* W_in = (const float*)d_in[1];
  const float* cw   = (const float*)d_in[2];
  const float* cb   = (const float*)d_in[3];
  const float* xpw  = (const float*)d_in[4];
  const float* dtw  = (const float*)d_in[5];
  const float* dtb  = (const float*)d_in[6];
  const float* alog = (const float*)d_in[7];
  const float* Dsk  = (const float*)d_in[8];
  const float* wout = (const float*)d_in[9];
  float* out = (float*)d_out;

  float* ws    = (float*)d_ws;
  float* xz    = ws;                                  // 16384*384
  float* xc    = xz + (size_t)MPIX * NXZ;             // 16384*192
  float* xdbl  = xc + (size_t)MPIX * DI;              // 16*4096*38
  float* y4    = xdbl + (size_t)16 * Lseq * CDBL;     // 4*16384*192
  float* yy    = y4 + (size_t)4 * MPIX * DI;          // 16384*192
  float* wt    = yy + (size_t)MPIX * DI;              // 9*192*192
  float* hprod = wt + (size_t)9 * DI * DI;            // 16*192*32*16
  float* hend  = hprod + (size_t)16 * DI * NCHUNK * DSt;
  float* hin   = hend + (size_t)16 * DI * NCHUNK * DSt;

  // Weight transpose (depends only on conv_w)
  k_wt<<<1296, 256, 0, stream>>>(cw, wt);
  // K1: input projection (1024*24 tiles, 8 waves/block)
  k_gemm_in<<<3072, 256, 0, stream>>>(x, W_in, xz);
  // K2: conv 3x3 + SiLU (1024*6 double tiles)
  k_conv<<<768, 256, 0, stream>>>(xz, wt, cb, xc);
  // K3: x_dbl projection (4096*3 tiles)
  k_xdbl<<<1536, 256, 0, stream>>>(xc, xpw, xdbl);
  // K4: chunked scan: local pass, serial fixup, corrected pass
  k_scan_chunk<0><<<16 * 12 * NCHUNK, 256, 0, stream>>>(
      xc, xdbl, alog, dtw, dtb, Dsk, hin, hprod, hend, y4);
  k_scan_mid<<<192, 256, 0, stream>>>(hprod, hend, hin);
  k_scan_chunk<1><<<16 * 12 * NCHUNK, 256, 0, stream>>>(
      xc, xdbl, alog, dtw, dtb, Dsk, hin, hprod, hend, y4);
  // K5: combine + gate
  k_combine<<<12288, 256, 0, stream>>>(y4, xz, yy);
  // K6: output projection (1024*6 tiles)
  k_gemm_out<<<768, 256, 0, stream>>>(yy, wout, out);
}